// global_local_fusion_28836410425556
// MI455X (gfx1250) — compile-verified
//
#include <hip/hip_runtime.h>
#include <hip/hip_bf16.h>

// ---------------------------------------------------------------------------
// Shapes: B=8, H=W=128, HW=16384, NPIX=131072
// f32 WMMA (V_WMMA_F32_16X16X4_F32) for all GEMM stages: full f32 precision,
// and the problem is HBM-bound (~30 GFLOP vs ~700 MB traffic) so f32 matrix
// throughput is more than sufficient on MI455X.
// ---------------------------------------------------------------------------

typedef float v2f __attribute__((ext_vector_type(2)));
typedef float v8f __attribute__((ext_vector_type(8)));

__device__ __forceinline__ v8f wmma_f32_k4(v2f a, v2f b, v8f c) {
  // 8 args: (neg_a, A, neg_b, B, c_mod, C, reuse_a, reuse_b)
  return __builtin_amdgcn_wmma_f32_16x16x4_f32(false, a, false, b, (short)0, c,
                                               false, false);
}

#define HW_   16384
#define NPIX_ 131072

// ---------------------------------------------------------------------------
// K1: cat = conv1x1(concat(pixel_shuffle(high), low); 128x256) + bias
//     GEMM M=128, K=256, N=NPIX. One block = 8 waves x 16 channels = all 128
//     rows for a 16-pixel tile. B-tile staged in LDS (on-the-fly concat).
// ---------------------------------------------------------------------------
__global__ void __launch_bounds__(256)
k_reduce(const float* __restrict__ hi,    // [8,128,64,64]
         const float* __restrict__ lo,    // [8,224,128,128]
         const float* __restrict__ rw,    // [128,256]
         const float* __restrict__ rb,    // [128]
         float* __restrict__ catR)        // [8,128,16384]
{
  __shared__ float lds[16 * 260];         // [n=16][k=256], stride 260 (pad)
  const int tid = threadIdx.x;
  const int pixBase = blockIdx.x * 16;

  // stage: thread t loads channel k = t/16 + 16*it of pixel n = t%16
  const int nS = tid & 15;
  const int kB = tid >> 4;
  const int pixS = pixBase + nS;
  const int bS  = pixS >> 14;
  const int hwS = pixS & (HW_ - 1);
  const int hS  = hwS >> 7;
  const int wS  = hwS & 127;
#pragma unroll
  for (int it = 0; it < 16; ++it) {
    int k = kB + 16 * it;
    float v;
    if (k < 32) {
      // pixel_shuffle(r=2): ps[b,k,h,w] = hi[b, 4k + 2*(h&1) + (w&1), h/2, w/2]
      int c = (k << 2) + ((hS & 1) << 1) + (wS & 1);
      v = hi[((size_t)bS * 128 + c) * 4096 + (size_t)(hS >> 1) * 64 + (wS >> 1)];
    } else {
      v = lo[((size_t)bS * 224 + (k - 32)) * HW_ + hwS];
    }
    lds[nS * 260 + k] = v;
  }
  __syncthreads();

  const int wid  = tid >> 5;
  const int lane = tid & 31;
  const int half = lane >> 4;
  const int l16  = lane & 15;
  const int mBase = wid * 16;
  const int m = mBase + l16;

  v8f acc;
#pragma unroll
  for (int v = 0; v < 8; ++v) acc[v] = rb[mBase + v + 8 * half];

#pragma unroll 4
  for (int kk = 0; kk < 64; ++kk) {
    int k0 = kk * 4 + 2 * half;
    v2f a = *(const v2f*)(rw + (size_t)m * 256 + k0);
    v2f b = *(const v2f*)(lds + l16 * 260 + k0);
    acc = wmma_f32_k4(a, b, acc);
  }

  const int pix = pixBase + l16;
  const int b0  = pix >> 14;
  const int hw  = pix & (HW_ - 1);
#pragma unroll
  for (int v = 0; v < 8; ++v) {
    int c = mBase + v + 8 * half;
    catR[((size_t)b0 * 128 + c) * HW_ + hw] = acc[v];
  }
}

// ---------------------------------------------------------------------------
// K2: mix = leaky( conv1x1(cat; 32x128) + conv1x1(emb; 32x256) )
//     Two WMMA K-loops summed into one accumulator. 8 waves: 2 mTiles x 4
//     pixel subtiles (64 px / block).
// ---------------------------------------------------------------------------
__global__ void __launch_bounds__(256)
k_mix(const float* __restrict__ catR,   // [8,128,16384]
      const float* __restrict__ emb,    // [8,256,16384]
      const float* __restrict__ w1,     // [32,128]
      const float* __restrict__ b1,     // [32]
      const float* __restrict__ we,     // [32,256]
      const float* __restrict__ be,     // [32]
      float* __restrict__ mixo)         // [8,32,16384]
{
  const int tid  = threadIdx.x;
  const int wid  = tid >> 5;
  const int lane = tid & 31;
  const int half = lane >> 4;
  const int l16  = lane & 15;
  const int mBase = (wid & 1) * 16;
  const int m = mBase + l16;
  const int pix = blockIdx.x * 64 + (wid >> 1) * 16 + l16;
  const int b0  = pix >> 14;
  const int hw  = pix & (HW_ - 1);
  const size_t cb = (size_t)b0 * 128 * HW_ + hw;
  const size_t eb = (size_t)b0 * 256 * HW_ + hw;

  v8f acc;
#pragma unroll
  for (int v = 0; v < 8; ++v) {
    int o = mBase + v + 8 * half;
    acc[v] = b1[o] + be[o];
  }

#pragma unroll 4
  for (int kk = 0; kk < 32; ++kk) {
    int k0 = kk * 4 + 2 * half;
    v2f a = *(const v2f*)(w1 + (size_t)m * 128 + k0);
    v2f b;
    b.x = catR[cb + (size_t)k0 * HW_];
    b.y = catR[cb + (size_t)(k0 + 1) * HW_];
    acc = wmma_f32_k4(a, b, acc);
  }
#pragma unroll 4
  for (int kk = 0; kk < 64; ++kk) {
    int k0 = kk * 4 + 2 * half;
    v2f a = *(const v2f*)(we + (size_t)m * 256 + k0);
    v2f b;
    b.x = emb[eb + (size_t)k0 * HW_];
    b.y = emb[eb + (size_t)(k0 + 1) * HW_];
    acc = wmma_f32_k4(a, b, acc);
  }

#pragma unroll
  for (int v = 0; v < 8; ++v) {
    int c = mBase + v + 8 * half;
    float x = acc[v];
    x = x > 0.f ? x : 0.01f * x;
    mixo[((size_t)b0 * 32 + c) * HW_ + hw] = x;
  }
}

// ---------------------------------------------------------------------------
// K3: involution weights = conv3x3(mix; 200x32x3x3) + bias, as implicit GEMM
//     M=200 (pad->208, wave-uniform guard), K=288, 16-px tile per wave.
// ---------------------------------------------------------------------------
__global__ void __launch_bounds__(128)
k_conv2(const float* __restrict__ mixo,  // [8,32,16384]
        const float* __restrict__ w2,    // [200,288] = [o][c*9+i*3+j]
        const float* __restrict__ b2,    // [200]
        float* __restrict__ wgt)         // [8,200,16384]
{
  const int tid  = threadIdx.x;
  const int wid  = tid >> 5;
  const int lane = tid & 31;
  const int half = lane >> 4;
  const int l16  = lane & 15;
  const int mTile = blockIdx.y * 4 + wid;     // wave-uniform
  if (mTile >= 13) return;                    // whole wave exits: EXEC stays full
  const int mBase = mTile * 16;
  const int m = mBase + l16;

  const int pix = blockIdx.x * 16 + l16;
  const int b0  = pix >> 14;
  const int hw  = pix & (HW_ - 1);
  const int h   = hw >> 7;
  const int w   = hw & 127;
  const size_t mb = (size_t)b0 * 32 * HW_;

  v8f acc;
#pragma unroll
  for (int v = 0; v < 8; ++v) {
    int o = mBase + v + 8 * half;
    acc[v] = (o < 200) ? b2[o] : 0.f;
  }

#pragma unroll 4
  for (int kk = 0; kk < 72; ++kk) {
    int k0 = kk * 4 + 2 * half;
    v2f a;
    if (m < 200) a = *(const v2f*)(w2 + (size_t)m * 288 + k0);
    else { a.x = 0.f; a.y = 0.f; }
    v2f b;
#pragma unroll
    for (int e = 0; e < 2; ++e) {
      int k = k0 + e;
      int c = k / 9;
      int r = k - c * 9;
      int i = r / 3;
      int j = r - i * 3;
      int hh = h + i - 1;
      int ww = w + j - 1;
      float val = 0.f;
      if (hh >= 0 && hh < 128 && ww >= 0 && ww < 128)
        val = mixo[mb + (size_t)c * HW_ + hh * 128 + ww];
      b[e] = val;
    }
    acc = wmma_f32_k4(a, b, acc);
  }

#pragma unroll
  for (int v = 0; v < 8; ++v) {
    int o = mBase + v + 8 * half;
    if (o < 200)
      wgt[((size_t)b0 * 200 + o) * HW_ + hw] = acc[v];
  }
}

// ---------------------------------------------------------------------------
// K4: involution (per-pixel 5x5 weights, 8 groups x 16 ch, pad=2). One thread
//     per pixel; 25 group weights held in registers across the 16 channels.
// ---------------------------------------------------------------------------
__global__ void __launch_bounds__(128)
k_invol(const float* __restrict__ catR,  // [8,128,16384]
        const float* __restrict__ wgt,   // [8,200,16384], o = g*25+i*5+j
        float* __restrict__ invo)        // [8,128,16384]
{
  const int tid = threadIdx.x;
  const int pix = blockIdx.x * 128 + tid;
  const int b0  = pix >> 14;
  const int hw  = pix & (HW_ - 1);
  const int h   = hw >> 7;
  const int w   = hw & 127;
  const size_t wb = (size_t)b0 * 200 * HW_ + hw;
  const size_t cb = (size_t)b0 * 128 * HW_;

  for (int g = 0; g < 8; ++g) {
    float wr[25];
#pragma unroll
    for (int r = 0; r < 25; ++r)
      wr[r] = wgt[wb + (size_t)(g * 25 + r) * HW_];
    for (int cc = 0; cc < 16; ++cc) {
      int c = g * 16 + cc;
      const float* xp = catR + cb + (size_t)c * HW_;
      float acc = 0.f;
#pragma unroll
      for (int i = 0; i < 5; ++i) {
        int hh = h + i - 2;
        if (hh < 0 || hh >= 128) continue;
#pragma unroll
        for (int j = 0; j < 5; ++j) {
          int ww = w + j - 2;
          if (ww < 0 || ww >= 128) continue;
          acc += wr[i * 5 + j] * xp[hh * 128 + ww];
        }
      }
      invo[cb + (size_t)c * HW_ + hw] = acc;
    }
  }
}

// ---------------------------------------------------------------------------
// K5: out = leaky( BN( conv1x1(inv; 64x128) + bias ) ). GEMM M=64, K=128 with
//     BN/leaky epilogue at the store. 8 waves: 4 mTiles x 2 pixel subtiles.
// ---------------------------------------------------------------------------
__global__ void __launch_bounds__(256)
k_out(const float* __restrict__ invo,  // [8,128,16384]
      const float* __restrict__ ow,    // [64,128]
      const float* __restrict__ ob,    // [64]
      const float* __restrict__ gam, const float* __restrict__ bet,
      const float* __restrict__ mu,  const float* __restrict__ var,
      float* __restrict__ out)         // [8,64,16384]
{
  const int tid  = threadIdx.x;
  const int wid  = tid >> 5;
  const int lane = tid & 31;
  const int half = lane >> 4;
  const int l16  = lane & 15;
  const int mBase = (wid & 3) * 16;
  const int m = mBase + l16;
  const int pix = blockIdx.x * 32 + (wid >> 2) * 16 + l16;
  const int b0  = pix >> 14;
  const int hw  = pix & (HW_ - 1);
  const size_t ib = (size_t)b0 * 128 * HW_ + hw;

  v8f acc;
#pragma unroll
  for (int v = 0; v < 8; ++v) acc[v] = 0.f;

#pragma unroll 4
  for (int kk = 0; kk < 32; ++kk) {
    int k0 = kk * 4 + 2 * half;
    v2f a = *(const v2f*)(ow + (size_t)m * 128 + k0);
    v2f b;
    b.x = invo[ib + (size_t)k0 * HW_];
    b.y = invo[ib + (size_t)(k0 + 1) * HW_];
    acc = wmma_f32_k4(a, b, acc);
  }

#pragma unroll
  for (int v = 0; v < 8; ++v) {
    int o = mBase + v + 8 * half;
    float scale = gam[o] / sqrtf(var[o] + 1e-5f);
    float x = (acc[v] + ob[o] - mu[o]) * scale + bet[o];
    x = x > 0.f ? x : 0.01f * x;
    out[((size_t)b0 * 64 + o) * HW_ + hw] = x;
  }
}

// ---------------------------------------------------------------------------
extern "C" void kernel_launch(void* const* d_in, const int* in_sizes, int n_in,
                              void* d_out, int out_size, void* d_ws, size_t ws_size,
                              hipStream_t stream) {
  const float* hi  = (const float*)d_in[0];   // high_level [8,128,64,64]
  const float* lo  = (const float*)d_in[1];   // low_level  [8,224,128,128]
  const float* emb = (const float*)d_in[2];   // emb        [8,256,128,128]
  const float* rw  = (const float*)d_in[3];   // reduce_w   [128,256,1,1]
  const float* rb  = (const float*)d_in[4];
  const float* w1  = (const float*)d_in[5];   // conv1_w    [32,128,1,1]
  const float* b1  = (const float*)d_in[6];
  const float* w2  = (const float*)d_in[7];   // conv2_w    [200,32,3,3]
  const float* b2  = (const float*)d_in[8];
  const float* we  = (const float*)d_in[9];   // embconv_w  [32,256,1,1]
  const float* be  = (const float*)d_in[10];
  const float* ow  = (const float*)d_in[11];  // out_w      [64,128,1,1]
  const float* ob  = (const float*)d_in[12];
  const float* gam = (const float*)d_in[13];
  const float* bet = (const float*)d_in[14];
  const float* mu  = (const float*)d_in[15];
  const float* var = (const float*)d_in[16];

  // workspace layout (floats): catR 16.7M | mix 4.2M | wgt 26.2M | inv 16.7M
  float* ws   = (float*)d_ws;
  float* catR = ws;                 // [8,128,16384]
  float* mixb = ws + 16777216;      // [8, 32,16384]
  float* wgt  = ws + 20971520;      // [8,200,16384]
  float* invb = ws + 47185920;      // [8,128,16384]   (~244 MB total)
  float* outp = (float*)d_out;      // [8, 64,16384]

  k_reduce<<<dim3(NPIX_ / 16), dim3(256), 0, stream>>>(hi, lo, rw, rb, catR);
  k_mix   <<<dim3(NPIX_ / 64), dim3(256), 0, stream>>>(catR, emb, w1, b1, we, be, mixb);
  k_conv2 <<<dim3(NPIX_ / 16, 4), dim3(128), 0, stream>>>(mixb, w2, b2, wgt);
  k_invol <<<dim3(NPIX_ / 128), dim3(128), 0, stream>>>(catR, wgt, invb);
  k_out   <<<dim3(NPIX_ / 32), dim3(256), 0, stream>>>(invb, ow, ob, gam, bet, mu, var, outp);
}